// ChempropLayer_9801115369511
// MI455X (gfx1250) — compile-verified
//
#include <hip/hip_runtime.h>
#include <hip/hip_bf16.h>

#define NV 50000
#define NE 500000
#define DH 128
#define LDST 132   // padded LDS row stride (floats): bank-conflict-free strided reads
#define NTILES (NE / 16)   // 31250 16-edge tiles

typedef __attribute__((ext_vector_type(2))) float v2f;
typedef __attribute__((ext_vector_type(8))) float v8f;

// ---------------------------------------------------------------------------
// Kernel 1: messages = relu(edge_feats); node_messages[dest] += messages
// One wave (32 lanes) per edge, float4 per lane, hardware f32 atomics into L2
// (node_messages = 25.6 MB, resident in the 192 MB L2).
// ---------------------------------------------------------------------------
__global__ void dmpnn_scatter(const float* __restrict__ edge_feats,
                              const long long* __restrict__ edge_index,
                              float* __restrict__ node_msgs) {
    const int wave = (blockIdx.x * blockDim.x + threadIdx.x) >> 5;  // edge id
    const int lane = threadIdx.x & 31;
    const long long dst = edge_index[(size_t)NE + wave];            // dest row

    const float4 v = *(const float4*)(edge_feats + (size_t)wave * DH + lane * 4);
    float r0 = fmaxf(v.x, 0.0f);
    float r1 = fmaxf(v.y, 0.0f);
    float r2 = fmaxf(v.z, 0.0f);
    float r3 = fmaxf(v.w, 0.0f);

    float* p = node_msgs + (size_t)dst * DH + lane * 4;
    asm volatile("global_atomic_add_f32 %0, %1, off scope:SCOPE_DEV" :: "v"(p + 0), "v"(r0) : "memory");
    asm volatile("global_atomic_add_f32 %0, %1, off scope:SCOPE_DEV" :: "v"(p + 1), "v"(r1) : "memory");
    asm volatile("global_atomic_add_f32 %0, %1, off scope:SCOPE_DEV" :: "v"(p + 2), "v"(r2) : "memory");
    asm volatile("global_atomic_add_f32 %0, %1, off scope:SCOPE_DEV" :: "v"(p + 3), "v"(r3) : "memory");
}

// ---------------------------------------------------------------------------
// Kernel 2: out = (node_messages[src] - relu(edge_feats[rev])) @ W^T + b
// 8 waves/block; each wave owns a 16-edge tile and computes its full 16x128
// output slab with v_wmma_f32_16x16x4_f32 (8 accumulators x 32 K-steps).
// W staged once in LDS (padded stride); per-wave A tile staged in LDS.
// ---------------------------------------------------------------------------
__global__ void dmpnn_gemm(const float* __restrict__ edge_feats,
                           const float* __restrict__ W,
                           const float* __restrict__ bias,
                           const long long* __restrict__ edge_index,
                           const long long* __restrict__ rev_index,
                           const float* __restrict__ node_msgs,
                           float* __restrict__ out) {
    extern __shared__ float smem[];
    float* wT  = smem;                  // 128 * 132 floats, row j, col k = W[j][k]
    float* bS  = wT + DH * LDST;        // 128 floats
    float* aBs = bS + DH;               // 8 waves * 16 * 132 floats

    const int tid  = threadIdx.x;
    const int lane = tid & 31;
    const int wave = tid >> 5;

    // ---- stage W (16384 floats) as 4096 float4, padded stride 132 ----
    for (int i = tid; i < (DH * DH) / 4; i += 256) {
        const int j = (i * 4) >> 7;        // row (output channel)
        const int k = (i * 4) & (DH - 1);  // col (k), multiple of 4 -> 16B aligned in LDS
        *(float4*)(wT + j * LDST + k) = *(const float4*)(W + i * 4);
    }
    if (tid < DH) bS[tid] = bias[tid];
    __syncthreads();

    // tile clamp instead of branch: keeps EXEC all-ones for WMMA (tail waves
    // just recompute tile NTILES-1; deterministic duplicate stores).
    int tile = blockIdx.x * 8 + wave;
    if (tile >= NTILES) tile = NTILES - 1;
    const long long e0 = (long long)tile * 16;

    float* aT = aBs + wave * 16 * LDST;

    // ---- stage A tile: 16 rows of edge_messages; 32 lanes x float4 per row ----
    for (int r = 0; r < 16; ++r) {
        const long long e  = e0 + r;
        const long long s  = edge_index[e];   // src node
        const long long rv = rev_index[e];    // reverse edge
        const float4 nm = *(const float4*)(node_msgs + (size_t)s * DH + lane * 4);
        const float4 ef = *(const float4*)(edge_feats + (size_t)rv * DH + lane * 4);
        float4 em;
        em.x = nm.x - fmaxf(ef.x, 0.0f);
        em.y = nm.y - fmaxf(ef.y, 0.0f);
        em.z = nm.z - fmaxf(ef.z, 0.0f);
        em.w = nm.w - fmaxf(ef.w, 0.0f);
        *(float4*)(aT + r * LDST + lane * 4) = em;
    }
    __builtin_amdgcn_wave_barrier();   // DS ops are in-order per wave; fence compiler

    const int half = lane >> 4;        // 0: K=k0,k0+1   1: K=k0+2,k0+3
    const int nn   = lane & 15;        // N within tile / M row for A reads

    // ---- init accumulators with bias (C/D layout: N = lane%16, same per VGPR) ----
    v8f acc[8];
#pragma unroll
    for (int nt = 0; nt < 8; ++nt) {
        const float bj = bS[nt * 16 + nn];
#pragma unroll
        for (int i = 0; i < 8; ++i) acc[nt][i] = bj;
    }

    // ---- K loop: 32 steps of K=4, 8 WMMAs per step ----
#pragma unroll 4
    for (int k0 = 0; k0 < DH; k0 += 4) {
        // A 16x4: lane m = lane%16 holds row m; x=K(k0+2*half), y=K(k0+2*half+1)
        const v2f a = *(const v2f*)(aT + nn * LDST + k0 + 2 * half);
#pragma unroll
        for (int nt = 0; nt < 8; ++nt) {
            // B 4x16: B[k][n] = W[n0+n][k]; x=K(k0+2*half), y=next K
            const v2f bb = *(const v2f*)(wT + (nt * 16 + nn) * LDST + k0 + 2 * half);
            acc[nt] = __builtin_amdgcn_wmma_f32_16x16x4_f32(
                false, a, false, bb, (short)0, acc[nt], false, false);
        }
    }

    // ---- store: VGPR i -> row (i + 8*half), col nt*16 + nn ----
#pragma unroll
    for (int nt = 0; nt < 8; ++nt) {
#pragma unroll
        for (int i = 0; i < 8; ++i) {
            const int m = i + 8 * half;
            out[(size_t)(e0 + m) * DH + nt * 16 + nn] = acc[nt][i];
        }
    }
}

// ---------------------------------------------------------------------------
// inputs: 0=edge_feats[E,128] f32, 1=node_feats[V,1] f32 (shape only),
//         2=W[128,128] f32, 3=b[128] f32, 4=edge_index[2,E] i64, 5=rev_index[E] i64
// out: [E,128] f32.  ws: node_messages [V,128] f32 = 25.6 MB.
// ---------------------------------------------------------------------------
extern "C" void kernel_launch(void* const* d_in, const int* in_sizes, int n_in,
                              void* d_out, int out_size, void* d_ws, size_t ws_size,
                              hipStream_t stream) {
    const float*     edge_feats = (const float*)d_in[0];
    const float*     W          = (const float*)d_in[2];
    const float*     bias       = (const float*)d_in[3];
    const long long* edge_index = (const long long*)d_in[4];
    const long long* rev_index  = (const long long*)d_in[5];
    float*           out        = (float*)d_out;
    float*           node_msgs  = (float*)d_ws;

    hipMemsetAsync(node_msgs, 0, (size_t)NV * DH * sizeof(float), stream);

    // one wave per edge -> 8 edges / 256-thread block; NE % 8 == 0
    dmpnn_scatter<<<NE / 8, 256, 0, stream>>>(edge_feats, edge_index, node_msgs);

    const size_t smem = (size_t)(DH * LDST + DH + 8 * 16 * LDST) * sizeof(float); // ~135 KB
    const int blocks = (NTILES + 7) / 8;   // 3907
    dmpnn_gemm<<<blocks, 256, smem, stream>>>(edge_feats, W, bias, edge_index,
                                              rev_index, node_msgs, out);
}